// Lag_PP_NN_5970004542130
// MI455X (gfx1250) — compile-verified
//
#include <hip/hip_runtime.h>

#define Bdim 8
#define Ldim 1024
#define Tdim 10
#define NT   32                   // 1024/32 tiles per dim
#define PAIRS (NT*(NT+1)/2)       // 528
#define SLOG 2.1972245773362196f  // log(9)

typedef float v2f __attribute__((ext_vector_type(2)));
typedef float v4f __attribute__((ext_vector_type(4)));
typedef float v8f __attribute__((ext_vector_type(8)));

__device__ __forceinline__ float sigm(float x) { return 1.0f / (1.0f + __expf(-x)); }
__device__ __forceinline__ float ssign(float x) { return sigm(2.0f * x); } // K=1

__device__ __forceinline__ float mlp3(const float* W1, const float* b1,
                                      const float* W2, float b2,
                                      float f0, float f1, float f2) {
  float h0 = fmaxf(0.f, fmaf(W1[0], f0, fmaf(W1[1], f1, fmaf(W1[2], f2, b1[0]))));
  float h1 = fmaxf(0.f, fmaf(W1[3], f0, fmaf(W1[4], f1, fmaf(W1[5], f2, b1[1]))));
  float h2 = fmaxf(0.f, fmaf(W1[6], f0, fmaf(W1[7], f1, fmaf(W1[8], f2, b1[2]))));
  return fmaxf(0.f, fmaf(W2[0], h0, fmaf(W2[1], h1, fmaf(W2[2], h2, b2))));
}

// m[i,j] from per-row features (ba,bu,bc,bg) = (x,y,z,w)
__device__ __forceinline__ float symm(const float4 xi, const float4 xj) {
  return xi.x*xj.y + xj.x*xi.y + xi.z*xj.w + xj.z*xi.w + xi.y*xj.w + xj.y*xi.w;
}

__device__ __forceinline__ void pair_decode(int p, int& ti, int& tj) {
  int t = 0, rem = p;
  while (rem >= NT - t) { rem -= NT - t; t++; }
  ti = t; tj = t + rem;
}

// Row-sum of a 16x16 (half of a 32-col tile) via V_WMMA_F32_16X16X4_F32 with
// an all-ones B matrix: D[M][N] = sum_k A[M,k]. fp32-exact (x*1.0 products).
// src points at row (rg*16); sumBase at the matching global row-sum slot.
__device__ __forceinline__ void wmma_rowsum_atomic(const float (*src)[33], int lane,
                                                   float* sumBase, int kh) {
#if __has_builtin(__builtin_amdgcn_wmma_f32_16x16x4_f32)
  // A (16x4 f32) layout: lanes 0-15 hold K=0,1 ; lanes 16-31 hold K=2,3 ; M = lane&15
  int M = lane & 15;
  int koff = (lane >> 4) * 2;
  v8f acc = {0.f, 0.f, 0.f, 0.f, 0.f, 0.f, 0.f, 0.f};
  v2f bv; bv.x = 1.f; bv.y = 1.f;   // all-ones B, layout-invariant
#pragma unroll
  for (int c = 0; c < 16; c += 4) {
    int cc = kh * 16 + c + koff;
    v2f av; av.x = src[M][cc]; av.y = src[M][cc + 1];
    acc = __builtin_amdgcn_wmma_f32_16x16x4_f32(false, av, false, bv,
                                                (short)0, acc, false, false);
  }
  // D layout: lane L<16 -> N=L, M=v ; lane>=16 -> N=L-16, M=v+8. Column N==0 holds sums.
  if ((lane & 15) == 0) {
    float* sp = sumBase + (lane >> 4) * 8;
#pragma unroll
    for (int v = 0; v < 8; v++) atomicAdd(sp + v, acc[v]);
  }
#else
  int M = lane & 15;
  if (lane < 16) {
    float s = 0.f;
    for (int c = 0; c < 16; c++) s += src[M][kh * 16 + c];
    atomicAdd(sumBase + M, s);
  }
#endif
}

// ---------------------------------------------------------------------------
// Init: uu = softsign(u-S)*u ; a_hat = sigmoid(uu)*softsign(uu-S) ;
// rowsum(contact(a_hat)) -> sum0 (atomics). Stores uu, ah to workspace.
// ---------------------------------------------------------------------------
__global__ void __launch_bounds__(256) init_kernel(
    const float* __restrict__ u, const float4* __restrict__ x4,
    float* __restrict__ uu, float* __restrict__ ah, float* __restrict__ sum0) {
  __shared__ float nA[32][33], nB[32][33];
  __shared__ float oA[32][33], oB[32][33];
  __shared__ float4 xA[32], xB[32];

  int b = blockIdx.y;
  int ti, tj; pair_decode(blockIdx.x, ti, tj);
  bool diag = (ti == tj);
  int tid = threadIdx.x;
  int r = tid >> 3, c0 = (tid & 7) * 4;
  size_t base = (size_t)b * Ldim * Ldim;
  size_t rowA = base + (size_t)(ti * 32 + r) * Ldim + tj * 32;
  size_t rowB = base + (size_t)(tj * 32 + r) * Ldim + ti * 32;

  if (tid < 32)       xA[tid] = x4[b * Ldim + ti * 32 + tid];
  else if (tid < 64)  xB[tid - 32] = x4[b * Ldim + tj * 32 + (tid - 32)];

  float4 uA4 = *(const float4*)(u + rowA + c0);
  float uAr[4] = {uA4.x, uA4.y, uA4.z, uA4.w};
  float uuAr[4], ahAr[4];
#pragma unroll
  for (int k = 0; k < 4; k++) {
    float uv = uAr[k];
    float uuv = ssign(uv - SLOG) * uv;
    float av = sigm(uuv) * ssign(uuv - SLOG);
    uuAr[k] = uuv; ahAr[k] = av; nA[r][c0 + k] = av;
  }
  *(float4*)(uu + rowA + c0) = make_float4(uuAr[0], uuAr[1], uuAr[2], uuAr[3]);
  *(float4*)(ah + rowA + c0) = make_float4(ahAr[0], ahAr[1], ahAr[2], ahAr[3]);

  if (!diag) {
    float4 uB4 = *(const float4*)(u + rowB + c0);
    float uBr[4] = {uB4.x, uB4.y, uB4.z, uB4.w};
    float uuBr[4], ahBr[4];
#pragma unroll
    for (int k = 0; k < 4; k++) {
      float uv = uBr[k];
      float uuv = ssign(uv - SLOG) * uv;
      float av = sigm(uuv) * ssign(uuv - SLOG);
      uuBr[k] = uuv; ahBr[k] = av; nB[r][c0 + k] = av;
    }
    *(float4*)(uu + rowB + c0) = make_float4(uuBr[0], uuBr[1], uuBr[2], uuBr[3]);
    *(float4*)(ah + rowB + c0) = make_float4(ahBr[0], ahBr[1], ahBr[2], ahBr[3]);
  }
  __syncthreads();

  float4 xiA = xA[r];
#pragma unroll
  for (int k = 0; k < 4; k++) {
    int c = c0 + k;
    float t1 = nA[r][c];
    float t2 = diag ? nA[c][r] : nB[c][r];
    oA[r][c] = 0.5f * (t1 * t1 + t2 * t2) * symm(xiA, xB[c]);
  }
  if (!diag) {
    float4 xiB = xB[r];
#pragma unroll
    for (int k = 0; k < 4; k++) {
      int c = c0 + k;
      float t1 = nB[r][c];
      float t2 = nA[c][r];
      oB[r][c] = 0.5f * (t1 * t1 + t2 * t2) * symm(xiB, xA[c]);
    }
  }
  __syncthreads();

  int wave = tid >> 5, lane = tid & 31;
  int bsel = wave & 1, rg = (wave >> 1) & 1, kh = (wave >> 2) & 1;
  if (!(diag && bsel)) {
    const float(*src)[33] = bsel ? (const float(*)[33]) & oB[rg * 16]
                                 : (const float(*)[33]) & oA[rg * 16];
    float* sb = sum0 + b * Ldim + (bsel ? tj : ti) * 32 + rg * 16;
    wmma_rowsum_atomic(src, lane, sb, kh);
  }
}

// ---------------------------------------------------------------------------
// One timestep: grad -> MLPs -> ah_new (in-place ah update) -> a = contact
// -> out[t] + partial rowsums into sumNext.
// ---------------------------------------------------------------------------
__global__ void __launch_bounds__(256) step_kernel(
    const float* __restrict__ uu, float* __restrict__ ah,
    const float4* __restrict__ x4, const float* __restrict__ cbuf,
    const float* __restrict__ aW1, const float* __restrict__ ab1,
    const float* __restrict__ aW2, const float* __restrict__ ab2,
    const float* __restrict__ rW1, const float* __restrict__ rb1,
    const float* __restrict__ rW2, const float* __restrict__ rb2,
    float* __restrict__ out, float* __restrict__ sumNext) {
  __shared__ float uuA[32][33], uuB[32][33];
  __shared__ float nA[32][33], nB[32][33];
  __shared__ float oA[32][33], oB[32][33];
  __shared__ float4 xA[32], xB[32];
  __shared__ float cA[32], cB[32];

  int b = blockIdx.y;
  int ti, tj; pair_decode(blockIdx.x, ti, tj);
  bool diag = (ti == tj);
  int tid = threadIdx.x;
  int r = tid >> 3, c0 = (tid & 7) * 4;
  size_t base = (size_t)b * Ldim * Ldim;
  size_t rowA = base + (size_t)(ti * 32 + r) * Ldim + tj * 32;
  size_t rowB = base + (size_t)(tj * 32 + r) * Ldim + ti * 32;

  float4 uA4 = *(const float4*)(uu + rowA + c0);
  float4 uB4 = *(const float4*)(uu + rowB + c0);
  float4 aA4 = *(const float4*)(ah + rowA + c0);
  float4 aB4 = *(const float4*)(ah + rowB + c0);
  uuA[r][c0] = uA4.x; uuA[r][c0 + 1] = uA4.y; uuA[r][c0 + 2] = uA4.z; uuA[r][c0 + 3] = uA4.w;
  uuB[r][c0] = uB4.x; uuB[r][c0 + 1] = uB4.y; uuB[r][c0 + 2] = uB4.z; uuB[r][c0 + 3] = uB4.w;
  if (tid < 32) {
    xA[tid] = x4[b * Ldim + ti * 32 + tid];
    cA[tid] = cbuf[b * Ldim + ti * 32 + tid];
  } else if (tid < 64) {
    int q = tid - 32;
    xB[q] = x4[b * Ldim + tj * 32 + q];
    cB[q] = cbuf[b * Ldim + tj * 32 + q];
  }
  __syncthreads();

  // Tiny MLP weights: uniform addresses -> scalar loads
  float w1a[9], b1a[3], w2a[3], b2a, w1r[9], b1r[3], w2r[3], b2r;
#pragma unroll
  for (int i = 0; i < 9; i++) { w1a[i] = aW1[i]; w1r[i] = rW1[i]; }
#pragma unroll
  for (int i = 0; i < 3; i++) {
    b1a[i] = ab1[i]; w2a[i] = aW2[i]; b1r[i] = rb1[i]; w2r[i] = rW2[i];
  }
  b2a = ab2[0]; b2r = rb2[0];

  float ahAr[4] = {aA4.x, aA4.y, aA4.z, aA4.w};
  float ahBr[4] = {aB4.x, aB4.y, aB4.z, aB4.w};
  float mA[4], mB[4], nAr[4], nBr[4];

  { // block A = [ti, tj]
    float4 xiA = xA[r];
    float ciA = cA[r];
#pragma unroll
    for (int k = 0; k < 4; k++) {
      int c = c0 + k;
      float uij = uuA[r][c], uji = uuB[c][r];
      float m = symm(xiA, xB[c]);
      mA[k] = m;
      float g = ahAr[k] * m * fmaf(-0.5f, uij + uji, ciA + cB[c]);
      float v = mlp3(w1a, b1a, w2a, b2a, ahAr[k], g, uij);
      float rho = mlp3(w1r, b1r, w2r, b2r, ahAr[k], g, uij);
      float ahn = fminf(fmaxf(fabsf(v) - rho, 0.f), 1.f);
      nAr[k] = ahn; nA[r][c] = ahn;
    }
    *(float4*)(ah + rowA + c0) = make_float4(nAr[0], nAr[1], nAr[2], nAr[3]);
  }
  if (!diag) { // block B = [tj, ti]
    float4 xiB = xB[r];
    float ciB = cB[r];
#pragma unroll
    for (int k = 0; k < 4; k++) {
      int c = c0 + k;
      float uij = uuB[r][c], uji = uuA[c][r];
      float m = symm(xiB, xA[c]);
      mB[k] = m;
      float g = ahBr[k] * m * fmaf(-0.5f, uij + uji, ciB + cA[c]);
      float v = mlp3(w1a, b1a, w2a, b2a, ahBr[k], g, uij);
      float rho = mlp3(w1r, b1r, w2r, b2r, ahBr[k], g, uij);
      float ahn = fminf(fmaxf(fabsf(v) - rho, 0.f), 1.f);
      nBr[k] = ahn; nB[r][c] = ahn;
    }
    *(float4*)(ah + rowB + c0) = make_float4(nBr[0], nBr[1], nBr[2], nBr[3]);
  }
  __syncthreads();

  { // a = 0.5*(ahn^2 + ahn^T^2) * m ; stream out with NT stores (no reuse)
    float oAr[4];
#pragma unroll
    for (int k = 0; k < 4; k++) {
      int c = c0 + k;
      float t1 = nA[r][c];
      float t2 = diag ? nA[c][r] : nB[c][r];
      float a = 0.5f * (t1 * t1 + t2 * t2) * mA[k];
      oAr[k] = a; oA[r][c] = a;
    }
    v4f oAst = {oAr[0], oAr[1], oAr[2], oAr[3]};
    __builtin_nontemporal_store(oAst, (v4f*)(out + rowA + c0));
    if (!diag) {
      float oBr[4];
#pragma unroll
      for (int k = 0; k < 4; k++) {
        int c = c0 + k;
        float t1 = nB[r][c];
        float t2 = nA[c][r];
        float a = 0.5f * (t1 * t1 + t2 * t2) * mB[k];
        oBr[k] = a; oB[r][c] = a;
      }
      v4f oBst = {oBr[0], oBr[1], oBr[2], oBr[3]};
      __builtin_nontemporal_store(oBst, (v4f*)(out + rowB + c0));
    }
  }
  __syncthreads();

  int wave = tid >> 5, lane = tid & 31;
  int bsel = wave & 1, rg = (wave >> 1) & 1, kh = (wave >> 2) & 1;
  if (!(diag && bsel)) {
    const float(*src)[33] = bsel ? (const float(*)[33]) & oB[rg * 16]
                                 : (const float(*)[33]) & oA[rg * 16];
    float* sb = sumNext + b * Ldim + (bsel ? tj : ti) * 32 + rg * 16;
    wmma_rowsum_atomic(src, lane, sb, kh);
  }
}

// ---------------------------------------------------------------------------
// Per-row: lambda update (2->3->1 MLP or init relu), c = lambda*softsign(s-1),
// and zero the other sum buffer for the next step.
// ---------------------------------------------------------------------------
__global__ void row_kernel(const float* __restrict__ sumCur, float* __restrict__ sumOther,
                           float* __restrict__ lmbd, float* __restrict__ cbuf,
                           const float* __restrict__ lW1, const float* __restrict__ lb1,
                           const float* __restrict__ lW2, const float* __restrict__ lb2,
                           int isInit) {
  int idx = blockIdx.x * blockDim.x + threadIdx.x;
  if (idx >= Bdim * Ldim) return;
  float s = sumCur[idx];
  float lm;
  if (isInit) {
    lm = fmaxf(0.f, s - 1.f); // W_PEN = 1
  } else {
    float lg = fmaxf(0.f, s - 1.f);
    float l0 = lmbd[idx];
    float h0 = fmaxf(0.f, fmaf(lW1[0], l0, fmaf(lW1[1], lg, lb1[0])));
    float h1 = fmaxf(0.f, fmaf(lW1[2], l0, fmaf(lW1[3], lg, lb1[1])));
    float h2 = fmaxf(0.f, fmaf(lW1[4], l0, fmaf(lW1[5], lg, lb1[2])));
    lm = fmaxf(0.f, fmaf(lW2[0], h0, fmaf(lW2[1], h1, fmaf(lW2[2], h2, lb2[0]))));
  }
  lmbd[idx] = lm;
  cbuf[idx] = lm * ssign(s - 1.f);
  sumOther[idx] = 0.f;
}

__global__ void zero_kernel(float* __restrict__ p, int n) {
  int i = blockIdx.x * blockDim.x + threadIdx.x;
  if (i < n) p[i] = 0.f;
}

extern "C" void kernel_launch(void* const* d_in, const int* in_sizes, int n_in,
                              void* d_out, int out_size, void* d_ws, size_t ws_size,
                              hipStream_t stream) {
  (void)in_sizes; (void)n_in; (void)out_size; (void)ws_size;
  const float* u   = (const float*)d_in[0];
  const float* x   = (const float*)d_in[1];
  // d_in[2] = timesteps (hardcoded Tdim)
  const float* aW1 = (const float*)d_in[3];
  const float* ab1 = (const float*)d_in[4];
  const float* aW2 = (const float*)d_in[5];
  const float* ab2 = (const float*)d_in[6];
  const float* rW1 = (const float*)d_in[7];
  const float* rb1 = (const float*)d_in[8];
  const float* rW2 = (const float*)d_in[9];
  const float* rb2 = (const float*)d_in[10];
  const float* lW1 = (const float*)d_in[11];
  const float* lb1 = (const float*)d_in[12];
  const float* lW2 = (const float*)d_in[13];
  const float* lb2 = (const float*)d_in[14];
  float* out = (float*)d_out;

  const size_t NLL = (size_t)Bdim * Ldim * Ldim;
  const int BL = Bdim * Ldim;
  float* uu   = (float*)d_ws;
  float* ahb  = uu + NLL;
  float* sum0 = ahb + NLL;
  float* sum1 = sum0 + BL;
  float* lmbd = sum1 + BL;
  float* cbuf = lmbd + BL;

  dim3 grid(PAIRS, Bdim), block(256);
  dim3 rgrid((BL + 255) / 256), rblock(256);

  zero_kernel<<<rgrid, rblock, 0, stream>>>(sum0, BL);
  init_kernel<<<grid, block, 0, stream>>>(u, (const float4*)x, uu, ahb, sum0);
  row_kernel<<<rgrid, rblock, 0, stream>>>(sum0, sum1, lmbd, cbuf, lW1, lb1, lW2, lb2, 1);

  for (int t = 0; t < Tdim; t++) {
    float* sumNext = (t & 1) ? sum0 : sum1;
    float* sumZero = (t & 1) ? sum1 : sum0;
    step_kernel<<<grid, block, 0, stream>>>(
        uu, ahb, (const float4*)x, cbuf,
        aW1 + 9 * t, ab1 + 3 * t, aW2 + 3 * t, ab2 + t,
        rW1 + 9 * t, rb1 + 3 * t, rW2 + 3 * t, rb2 + t,
        out + (size_t)t * NLL, sumNext);
    row_kernel<<<rgrid, rblock, 0, stream>>>(sumNext, sumZero, lmbd, cbuf,
        lW1 + 6 * t, lb1 + 3 * t, lW2 + 3 * t, lb2 + t, 0);
  }
}